// ModuleSoftsplat_47141561041514
// MI455X (gfx1250) — compile-verified
//
#include <hip/hip_runtime.h>
#include <stdint.h>

#ifndef __has_builtin
#define __has_builtin(x) 0
#endif

// Problem dims (fixed by reference): inp [8,64,256,448] f32, flow [8,2,256,448] f32
namespace {
constexpr int Bc = 8, Cc = 64, Hc = 256, Wc = 448;
constexpr int HWc = Hc * Wc;           // 114688
constexpr int TW = 32, TH = 8;         // source tile per block (256 px)
constexpr int PAD = 4;                 // halo; |flow|<4 covered (~99.97% of N(0,1))
constexpr int WW = TW + 2 * PAD;       // 40
constexpr int WH = TH + 2 * PAD;       // 16
constexpr int WN = WW * WH;            // 640 window cells
constexpr int CC = 8;                  // channels per pass
constexpr int NT = TW * TH;            // 256 threads / block (8 waves, wave32)
constexpr int NCHUNK = Cc / CC;        // 8
}

// Async-to-LDS builtin pointee types (from hipcc diagnostics):
//   b32:  int*            (AS1 src, AS3 dst)
//   b128: int __vector_size__(16)*  (AS1 src, AS3 dst)
typedef int v4i __attribute__((vector_size(16)));
typedef __attribute__((address_space(1))) int* as1_ip;
typedef __attribute__((address_space(3))) int* as3_ip;
typedef __attribute__((address_space(1))) v4i* as1_v4ip;
typedef __attribute__((address_space(3))) v4i* as3_v4ip;

#if __has_builtin(__builtin_amdgcn_global_load_async_to_lds_b128)
#define SOFTSPLAT_ASYNC 1
#define SOFTSPLAT_ASYNC_B128 1
#elif __has_builtin(__builtin_amdgcn_global_load_async_to_lds_b32)
#define SOFTSPLAT_ASYNC 1
#define SOFTSPLAT_ASYNC_B128 0
#else
#define SOFTSPLAT_ASYNC 0
#define SOFTSPLAT_ASYNC_B128 0
#endif

// Stage one CC-channel tile chunk (CC*NT f32 words) from global to LDS.
// blockBase points at element (b, cc, sy0, sx0); ldsBuf is the NT-per-channel
// staging buffer laid out as [ch][py*TW+px] == flat index f.
__device__ __forceinline__ void stage_chunk(const float* blockBase, float* ldsBuf,
                                            int tid) {
#if SOFTSPLAT_ASYNC_B128
  // 512 x B128 DMA ops (16 B per lane-op), 2 per thread. Rows are contiguous
  // (TW=32 px) and 16B-aligned (W*4 and sx0*4 are multiples of 16).
#pragma unroll
  for (int i = 0; i < (CC * NT) / (4 * NT); ++i) {  // 2 iterations
    const int f = 4 * (tid + NT * i);
    const int ch = f >> 8;            // f / NT
    const int pix = f & (NT - 1);
    const int py = pix >> 5;
    const int px = pix & (TW - 1);
    const float* gp = blockBase + (size_t)ch * HWc + py * Wc + px;
    __builtin_amdgcn_global_load_async_to_lds_b128(
        (as1_v4ip)(uintptr_t)gp, (as3_v4ip)(uint32_t)(uintptr_t)(ldsBuf + f), 0, 0);
  }
#elif SOFTSPLAT_ASYNC
#pragma unroll
  for (int ch = 0; ch < CC; ++ch) {
    const int f = ch * NT + tid;
    const int py = tid >> 5;
    const int px = tid & (TW - 1);
    const float* gp = blockBase + (size_t)ch * HWc + py * Wc + px;
    __builtin_amdgcn_global_load_async_to_lds_b32(
        (as1_ip)(uintptr_t)gp, (as3_ip)(uint32_t)(uintptr_t)(ldsBuf + f), 0, 0);
  }
#else
#pragma unroll
  for (int ch = 0; ch < CC; ++ch) {
    const int py = tid >> 5;
    const int px = tid & (TW - 1);
    ldsBuf[ch * NT + tid] = blockBase[(size_t)ch * HWc + py * Wc + px];
  }
#endif
}

__device__ __forceinline__ void stage_wait_all() {
#if SOFTSPLAT_ASYNC
#if __has_builtin(__builtin_amdgcn_s_wait_asynccnt)
  __builtin_amdgcn_s_wait_asynccnt(0);
#else
  asm volatile("s_wait_asynccnt 0" ::: "memory");
#endif
  asm volatile("" ::: "memory");  // keep LDS reads below the wait
#endif
}

__global__ void __launch_bounds__(NT) softsplat_zero_kernel(float4* __restrict__ out,
                                                            int n4) {
  int i = blockIdx.x * NT + threadIdx.x;
  int stride = gridDim.x * NT;
  float4 z = make_float4(0.f, 0.f, 0.f, 0.f);
  for (; i < n4; i += stride) out[i] = z;
}

__global__ void __launch_bounds__(NT) softsplat_splat_kernel(
    const float* __restrict__ inp, const float* __restrict__ flow,
    float* __restrict__ out) {
  __shared__ alignas(16) float shAcc[CC * WN];    // 20 KB accumulation window
  __shared__ alignas(16) float shIn[2][CC * NT];  // 16 KB double-buffered stage

  const int tid = threadIdx.x;
  const int tx = tid & (TW - 1);
  const int ty = tid >> 5;
  const int b = blockIdx.z;
  const int sx0 = blockIdx.x * TW;
  const int sy0 = blockIdx.y * TH;
  const int x = sx0 + tx;
  const int y = sy0 + ty;

  // ---- per-pixel flow -> 4 corner targets + weights (computed once) ----
  const size_t fbase = (size_t)b * 2 * HWc + (size_t)y * Wc + x;
  const float dx = flow[fbase];
  const float dy = flow[fbase + HWc];
  const float ox = (float)x + dx;
  const float oy = (float)y + dy;
  const float x0f = floorf(ox);
  const float y0f = floorf(oy);
  const float wx1 = ox - x0f, wy1 = oy - y0f;
  const float wx0 = 1.f - wx1, wy0 = 1.f - wy1;
  const int x0 = (int)x0f;
  const int y0 = (int)y0f;

  const int wx_org = sx0 - PAD;
  const int wy_org = sy0 - PAD;

  int ldsIdx[4], gIdx[4];
  float wgt[4];
  {
    const int cxs[4] = {x0, x0 + 1, x0, x0 + 1};
    const int cys[4] = {y0, y0, y0 + 1, y0 + 1};
    const float ws[4] = {wx0 * wy0, wx1 * wy0, wx0 * wy1, wx1 * wy1};
#pragma unroll
    for (int k = 0; k < 4; ++k) {
      const int cx = cxs[k], cy = cys[k];
      const bool valid = (cx >= 0) & (cx < Wc) & (cy >= 0) & (cy < Hc);
      const int lx = cx - wx_org, ly = cy - wy_org;
      const bool inWin = (lx >= 0) & (lx < WW) & (ly >= 0) & (ly < WH);
      wgt[k] = ws[k];
      ldsIdx[k] = (valid && inWin) ? (ly * WW + lx) : -1;   // LDS-atomic target
      gIdx[k] = (valid && !inWin) ? (cy * Wc + cx) : -1;    // global spill target
    }
  }

  // Block-level base: element (b, 0, sy0, sx0)
  const float* const blockBase = inp + (size_t)b * Cc * HWc + (size_t)sy0 * Wc + sx0;
  float* const outB = out + (size_t)b * Cc * HWc;  // + c*HWc + pixel

  // ---- prologue: stage chunk 0 ----
  stage_chunk(blockBase, &shIn[0][0], tid);

  for (int c = 0; c < NCHUNK; ++c) {
    const int buf = c & 1;

    // zero the accumulation window (B128 DS stores; overlaps in-flight DMA)
    {
      float4* const a4 = (float4*)shAcc;
#pragma unroll
      for (int i = tid; i < (CC * WN) / 4; i += NT) a4[i] = make_float4(0.f, 0.f, 0.f, 0.f);
    }

    stage_wait_all();   // this wave's async DMA complete...
    __syncthreads();    // ...then all waves' staging + zeroing visible

    // accumulate: 4 bilinear corners per element; LDS atomics for the window,
    // global hardware f32 atomics for rare far spills
#pragma unroll
    for (int ch = 0; ch < CC; ++ch) {
      const float v = shIn[buf][ch * NT + tid];
      float* const accCh = &shAcc[ch * WN];
      float* const outCh = outB + (size_t)(c * CC + ch) * HWc;
#pragma unroll
      for (int k = 0; k < 4; ++k) {
        const float contrib = v * wgt[k];
        if (ldsIdx[k] >= 0) {
          atomicAdd(&accCh[ldsIdx[k]], contrib);          // ds_add_f32
        } else if (gIdx[k] >= 0) {
          unsafeAtomicAdd(outCh + gIdx[k], contrib);      // global_atomic_add_f32
        }
      }
    }

    // kick off async stage of the next chunk; it overlaps the flush below
    if (c + 1 < NCHUNK)
      stage_chunk(blockBase + (size_t)(c + 1) * CC * HWc, &shIn[buf ^ 1][0], tid);
    __syncthreads();  // all ds_add done before flush reads the window

    // flush window to global; skip untouched (zero) halo cells
    for (int i = tid; i < CC * WN; i += NT) {
      const int ch = i / WN;
      const int cell = i - ch * WN;
      const int ly = cell / WW;
      const int lx = cell - ly * WW;
      const int gy = wy_org + ly;
      const int gx = wx_org + lx;
      if (gx >= 0 && gx < Wc && gy >= 0 && gy < Hc) {
        const float v = shAcc[i];
        if (v != 0.f)
          unsafeAtomicAdd(outB + (size_t)(c * CC + ch) * HWc + (size_t)gy * Wc + gx, v);
      }
    }
    __syncthreads();  // window may be re-zeroed next iteration
  }
}

extern "C" void kernel_launch(void* const* d_in, const int* in_sizes, int n_in,
                              void* d_out, int out_size, void* d_ws, size_t ws_size,
                              hipStream_t stream) {
  (void)in_sizes; (void)n_in; (void)d_ws; (void)ws_size;
  const float* inp = (const float*)d_in[0];
  const float* flw = (const float*)d_in[1];
  float* out = (float*)d_out;

  // d_out is poisoned by the harness and we accumulate atomically: zero first.
  const int n4 = out_size / 4;  // out_size = 58,720,256, divisible by 4
  softsplat_zero_kernel<<<8192, NT, 0, stream>>>((float4*)out, n4);

  dim3 grid(Wc / TW, Hc / TH, Bc);  // (14, 32, 8)
  softsplat_splat_kernel<<<grid, NT, 0, stream>>>(inp, flw, out);
}